// GAT_34772055228552
// MI455X (gfx1250) — compile-verified
//
#include <hip/hip_runtime.h>
#include <math.h>
#include <cstdint>

// ---------------------------------------------------------------------------
// 2-layer GAT for MI455X (gfx1250, wave32).
//   GEMM1: x[N,512] @ W1[512,32]  -> f32 WMMA 16x16x4, W1 staged in LDS
//           (K-pair-interleaved LDS layout: B fragment = one ds_load_b64)
//   edge phase: atomic segment-max / segment-sum / alpha-weighted scatter-add
//   GEMM2: h[N,32] @ W2[32,8]     -> f32 WMMA 16x16x4 (B zero-padded to 16)
// ---------------------------------------------------------------------------

#define N_NODES   100000
#define N_EDGES   3200000
#define FIN       512
#define FH        32          // H1*C1
#define H1        4
#define C1        8
#define C2        8
#define NEG_SLOPE 0.2f

typedef float v2f __attribute__((ext_vector_type(2)));
typedef float v8f __attribute__((ext_vector_type(8)));

// float atomic max via signed/unsigned integer ordering trick
// (valid with -inf initialization; works across mixed signs)
__device__ __forceinline__ void atomic_max_f32(float* addr, float val) {
  if (val >= 0.0f) atomicMax((int*)addr, __float_as_int(val));
  else             atomicMin((unsigned int*)addr, __float_as_uint(val));
}

// ---------------------------------------------------------------------------
__global__ __launch_bounds__(256) void fill_kernel(float* __restrict__ p,
                                                   float v, long n) {
  long i = (long)blockIdx.x * blockDim.x + threadIdx.x;
  long stride = (long)gridDim.x * blockDim.x;
  for (; i < n; i += stride) p[i] = v;
}

// ---------------------------------------------------------------------------
// GEMM1: xp[N,32] = x[N,512] @ W[512,32].  One wave per 16-row tile, two
// 16-col accumulators.  K-step 4 with V_WMMA_F32_16X16X4_F32.
// A layout (ISA 16x4 f32): lane&15 = M row; lane>>4 selects K pair {0,1}/{2,3}.
// B fragment for lane = (W[kb][col], W[kb+1][col]); LDS holds W in
// K-pair-interleaved form Wp[(k>>1)*64 + col*2 + (k&1)] so that fragment
// is one aligned ds_load_b64 (no register shuffles feeding the WMMA).
// C/D layout: VGPR v, lanes0-15 -> M=v, lanes16-31 -> M=v+8; N = lane&15.
__global__ __launch_bounds__(256) void gemm1_wmma(const float* __restrict__ x,
                                                  const float* __restrict__ W,
                                                  float* __restrict__ xp,
                                                  int ntiles) {
  __shared__ float Wp[FIN * FH];              // 64 KB of 320 KB WGP LDS
  for (int idx = threadIdx.x; idx < FIN * FH; idx += 256) {
    int k = idx >> 5, col = idx & 31;         // coalesced global read
    Wp[(k >> 1) * 64 + col * 2 + (k & 1)] = W[idx];
  }
  __syncthreads();

  const int wave = threadIdx.x >> 5;
  const int lane = threadIdx.x & 31;
  const int tile = blockIdx.x * 8 + wave;
  if (tile >= ntiles) return;                 // uniform per wave: EXEC stays ~0/~full

  const int mrow = lane & 15;
  const int ksel = lane >> 4;                 // 0 or 1
  const int col  = lane & 15;

  v8f c0 = {}, c1 = {};
  const float* xrow = x + (size_t)(tile * 16 + mrow) * FIN + 2 * ksel;
  const float* wbase = Wp + ksel * 64 + col * 2;

  for (int k = 0; k < FIN; k += 4) {
    v2f a  = *(const v2f*)(xrow + k);                    // global_load_b64
    v2f b0 = *(const v2f*)(wbase + (k >> 1) * 64);       // ds_load_b64
    v2f b1 = *(const v2f*)(wbase + (k >> 1) * 64 + 32);  // ds_load_b64
    c0 = __builtin_amdgcn_wmma_f32_16x16x4_f32(false, a, false, b0, (short)0, c0, false, false);
    c1 = __builtin_amdgcn_wmma_f32_16x16x4_f32(false, a, false, b1, (short)0, c1, false, false);
  }

  const int rbase = tile * 16 + 8 * ksel;
#pragma unroll
  for (int v = 0; v < 8; ++v) {
    xp[(size_t)(rbase + v) * FH + col]      = c0[v];
    xp[(size_t)(rbase + v) * FH + col + 16] = c1[v];
  }
}

// ---------------------------------------------------------------------------
// GEMM2: xp2[N,8] = h[N,32] @ W2[32,8].  B padded with zero cols 8..15,
// stored K-pair-interleaved: Wp[(k>>1)*32 + col*2 + (k&1)].
__global__ __launch_bounds__(256) void gemm2_wmma(const float* __restrict__ h,
                                                  const float* __restrict__ W,
                                                  float* __restrict__ xp2,
                                                  int ntiles) {
  __shared__ float Wp[FH * 16];
  for (int idx = threadIdx.x; idx < FH * 16; idx += 256) {
    int k = idx >> 4, cc = idx & 15;
    Wp[(k >> 1) * 32 + cc * 2 + (k & 1)] = (cc < C2) ? W[k * C2 + cc] : 0.0f;
  }
  __syncthreads();

  const int wave = threadIdx.x >> 5;
  const int lane = threadIdx.x & 31;
  const int tile = blockIdx.x * 8 + wave;
  if (tile >= ntiles) return;

  const int mrow = lane & 15;
  const int ksel = lane >> 4;
  const int col  = lane & 15;

  v8f c0 = {};
  const float* hrow  = h + (size_t)(tile * 16 + mrow) * FH + 2 * ksel;
  const float* wbase = Wp + ksel * 32 + col * 2;

#pragma unroll
  for (int k = 0; k < FH; k += 4) {
    v2f a  = *(const v2f*)(hrow + k);
    v2f b0 = *(const v2f*)(wbase + (k >> 1) * 32);
    c0 = __builtin_amdgcn_wmma_f32_16x16x4_f32(false, a, false, b0, (short)0, c0, false, false);
  }

  if (col < C2) {
    const int rbase = tile * 16 + 8 * ksel;
#pragma unroll
    for (int v = 0; v < 8; ++v)
      xp2[(size_t)(rbase + v) * C2 + col] = c0[v];
  }
}

// ---------------------------------------------------------------------------
// a_s[n,h] = sum_c xp[n,h,c]*att_src[h,c];  a_d likewise.
__global__ __launch_bounds__(256) void att_dots_kernel(
    const float* __restrict__ xp, const float* __restrict__ att_s,
    const float* __restrict__ att_d, float* __restrict__ as_,
    float* __restrict__ ad_, int H, int C) {
  int n = blockIdx.x * blockDim.x + threadIdx.x;
  if (n >= N_NODES) return;
  for (int hh = 0; hh < H; ++hh) {
    float s = 0.0f, d = 0.0f;
    for (int c = 0; c < C; ++c) {
      float v = xp[((size_t)n * H + hh) * C + c];
      s += v * att_s[hh * C + c];
      d += v * att_d[hh * C + c];
    }
    as_[n * H + hh] = s;
    ad_[n * H + hh] = d;
  }
}

// ---------------------------------------------------------------------------
// edge e in [0, E+N): e<E from edge_index, else self-loop (e-E, e-E).
__device__ __forceinline__ void edge_endpoints(const int* ei, int e, int& s, int& d) {
  if (e < N_EDGES) { s = ei[e]; d = ei[N_EDGES + e]; }
  else             { s = e - N_EDGES; d = s; }
}

__global__ __launch_bounds__(256) void edge_max_kernel(
    const int* __restrict__ ei, const float* __restrict__ as_,
    const float* __restrict__ ad_, float* __restrict__ amax, int H) {
  int e = blockIdx.x * blockDim.x + threadIdx.x;
  if (e >= N_EDGES + N_NODES) return;
  int s, d; edge_endpoints(ei, e, s, d);
  for (int hh = 0; hh < H; ++hh) {
    float v = as_[s * H + hh] + ad_[d * H + hh];
    v = (v > 0.0f) ? v : NEG_SLOPE * v;
    atomic_max_f32(&amax[d * H + hh], v);
  }
}

__global__ __launch_bounds__(256) void edge_sum_kernel(
    const int* __restrict__ ei, const float* __restrict__ as_,
    const float* __restrict__ ad_, const float* __restrict__ amax,
    float* __restrict__ den, int H) {
  int e = blockIdx.x * blockDim.x + threadIdx.x;
  if (e >= N_EDGES + N_NODES) return;
  int s, d; edge_endpoints(ei, e, s, d);
  for (int hh = 0; hh < H; ++hh) {
    float v = as_[s * H + hh] + ad_[d * H + hh];
    v = (v > 0.0f) ? v : NEG_SLOPE * v;
    atomicAdd(&den[d * H + hh], __expf(v - amax[d * H + hh]));
  }
}

// one thread per (edge, head); recompute alpha, scatter-add C-channel message
__global__ __launch_bounds__(256) void edge_agg_kernel(
    const int* __restrict__ ei, const float* __restrict__ as_,
    const float* __restrict__ ad_, const float* __restrict__ amax,
    const float* __restrict__ den, const float* __restrict__ xp,
    float* __restrict__ out, int H, int C) {
  long t = (long)blockIdx.x * blockDim.x + threadIdx.x;
  long total = (long)(N_EDGES + N_NODES) * H;
  if (t >= total) return;
  int e  = (int)(t / H);
  int hh = (int)(t % H);
  int s, d; edge_endpoints(ei, e, s, d);
  float v = as_[s * H + hh] + ad_[d * H + hh];
  v = (v > 0.0f) ? v : NEG_SLOPE * v;
  float alpha = __expf(v - amax[d * H + hh]) / den[d * H + hh];
  const float* xs = xp + ((size_t)s * H + hh) * C;
  float* od = out + ((size_t)d * H + hh) * C;
  for (int c = 0; c < C; ++c) atomicAdd(&od[c], xs[c] * alpha);
}

// ---------------------------------------------------------------------------
__global__ __launch_bounds__(256) void bias_elu_kernel(float* __restrict__ h,
                                                       const float* __restrict__ b,
                                                       long n, int F) {
  long i = (long)blockIdx.x * blockDim.x + threadIdx.x;
  if (i >= n) return;
  float v = h[i] + b[i % F];
  h[i] = (v > 0.0f) ? v : (__expf(v) - 1.0f);
}

__global__ __launch_bounds__(256) void bias_add_kernel(float* __restrict__ o,
                                                       const float* __restrict__ b,
                                                       long n, int F) {
  long i = (long)blockIdx.x * blockDim.x + threadIdx.x;
  if (i >= n) return;
  o[i] += b[i % F];
}

// ---------------------------------------------------------------------------
static inline int cdiv(long a, long b) { return (int)((a + b - 1) / b); }

extern "C" void kernel_launch(void* const* d_in, const int* in_sizes, int n_in,
                              void* d_out, int out_size, void* d_ws, size_t ws_size,
                              hipStream_t stream) {
  const float* x    = (const float*)d_in[0];
  const int*   ei   = (const int*)  d_in[1];
  const float* W1   = (const float*)d_in[2];
  const float* as1w = (const float*)d_in[3];
  const float* ad1w = (const float*)d_in[4];
  const float* b1   = (const float*)d_in[5];
  const float* W2   = (const float*)d_in[6];
  const float* as2w = (const float*)d_in[7];
  const float* ad2w = (const float*)d_in[8];
  const float* b2   = (const float*)d_in[9];
  float* out = (float*)d_out;

  // workspace carve-up (floats)
  float* ws    = (float*)d_ws;
  float* xp1   = ws;                                  // N*32
  float* acc1  = xp1   + (size_t)N_NODES * FH;        // N*32 (aggregate -> h)
  float* as1   = acc1  + (size_t)N_NODES * FH;        // N*4
  float* ad1   = as1   + (size_t)N_NODES * H1;        // N*4
  float* amax1 = ad1   + (size_t)N_NODES * H1;        // N*4
  float* den1  = amax1 + (size_t)N_NODES * H1;        // N*4
  float* xp2   = den1  + (size_t)N_NODES * H1;        // N*8
  float* as2   = xp2   + (size_t)N_NODES * C2;        // N
  float* ad2   = as2   + (size_t)N_NODES;             // N
  float* amax2 = ad2   + (size_t)N_NODES;             // N
  float* den2  = amax2 + (size_t)N_NODES;             // N

  const long ETOT   = (long)N_EDGES + N_NODES;
  const int  ntiles = N_NODES / 16;                   // 6250, exact
  const int  gemmBlocks = cdiv(ntiles, 8);

  // ----- per-launch init (harness does not re-zero) -----
  fill_kernel<<<1024, 256, 0, stream>>>(acc1, 0.0f, (long)N_NODES * FH);
  fill_kernel<<<256,  256, 0, stream>>>(amax1, -INFINITY, (long)N_NODES * H1);
  fill_kernel<<<256,  256, 0, stream>>>(den1, 0.0f, (long)N_NODES * H1);
  fill_kernel<<<256,  256, 0, stream>>>(amax2, -INFINITY, (long)N_NODES);
  fill_kernel<<<256,  256, 0, stream>>>(den2, 0.0f, (long)N_NODES);
  fill_kernel<<<256,  256, 0, stream>>>(out, 0.0f, (long)N_NODES * C2);

  // ----- layer 1 -----
  gemm1_wmma<<<gemmBlocks, 256, 0, stream>>>(x, W1, xp1, ntiles);
  att_dots_kernel<<<cdiv(N_NODES, 256), 256, 0, stream>>>(xp1, as1w, ad1w, as1, ad1, H1, C1);
  edge_max_kernel<<<cdiv(ETOT, 256), 256, 0, stream>>>(ei, as1, ad1, amax1, H1);
  edge_sum_kernel<<<cdiv(ETOT, 256), 256, 0, stream>>>(ei, as1, ad1, amax1, den1, H1);
  edge_agg_kernel<<<cdiv(ETOT * H1, 256), 256, 0, stream>>>(ei, as1, ad1, amax1, den1,
                                                            xp1, acc1, H1, C1);
  bias_elu_kernel<<<cdiv((long)N_NODES * FH, 256), 256, 0, stream>>>(acc1, b1,
                                                                     (long)N_NODES * FH, FH);

  // ----- layer 2 -----
  gemm2_wmma<<<gemmBlocks, 256, 0, stream>>>(acc1, W2, xp2, ntiles);
  att_dots_kernel<<<cdiv(N_NODES, 256), 256, 0, stream>>>(xp2, as2w, ad2w, as2, ad2, 1, C2);
  edge_max_kernel<<<cdiv(ETOT, 256), 256, 0, stream>>>(ei, as2, ad2, amax2, 1);
  edge_sum_kernel<<<cdiv(ETOT, 256), 256, 0, stream>>>(ei, as2, ad2, amax2, den2, 1);
  edge_agg_kernel<<<cdiv(ETOT, 256), 256, 0, stream>>>(ei, as2, ad2, amax2, den2,
                                                       xp2, out, 1, C2);
  bias_add_kernel<<<cdiv((long)N_NODES * C2, 256), 256, 0, stream>>>(out, b2,
                                                                     (long)N_NODES * C2, C2);
}